// Net_GCNWeight_11227044511898
// MI455X (gfx1250) — compile-verified
//
#include <hip/hip_runtime.h>
#include <math.h>

// ---------------------------------------------------------------------------
// CDNA5 (gfx1250) GCN forward pass, fused:
//   3x [ GEMM(fp32 WMMA, fused prev-layer bias+relu on load, fused self-loop
//        init on store) -> edge scatter (L2-resident f32 atomics) ]
//   -> sorted-batch mean pool (fused bias+relu) -> MLP head + log_softmax
// ---------------------------------------------------------------------------

typedef __attribute__((ext_vector_type(2))) float v2f;
typedef __attribute__((ext_vector_type(8))) float v8f;
typedef __attribute__((vector_size(4 * sizeof(int)))) int v4i;

#define NFEAT   128
#define NGRAPH  128
#define NCLS    10

#if defined(__has_builtin)
#if __has_builtin(__builtin_amdgcn_global_load_async_to_lds_b128) && \
    __has_builtin(__builtin_amdgcn_s_wait_asynccnt)
#define ASYNC_LDS_COPY 1
#endif
#endif

// ---------------------------------------------------------------------------
// deg[dst] += ew   (self-loop "+1" folded into rsqrt kernel below)
__global__ void deg_kernel(const int* __restrict__ dst,
                           const float* __restrict__ ew,
                           float* __restrict__ deg, int nEdges) {
    int t = blockIdx.x * blockDim.x + threadIdx.x;
    if (t < nEdges) atomicAdd(&deg[dst[t]], ew[t]);
}

// dinv = rsqrt(deg + 1)   (deg+1 >= 1 always, matching reference's guarded rsqrt)
__global__ void rsqrt_kernel(float* __restrict__ deg, int n) {
    int t = blockIdx.x * blockDim.x + threadIdx.x;
    if (t < n) deg[t] = rsqrtf(deg[t] + 1.0f);
}

// norm[e] = dinv[src] * ew * dinv[dst]
__global__ void norm_kernel(const int* __restrict__ src, const int* __restrict__ dst,
                            const float* __restrict__ ew, const float* __restrict__ dinv,
                            float* __restrict__ norm, int nEdges) {
    int t = blockIdx.x * blockDim.x + threadIdx.x;
    if (t < nEdges) norm[t] = dinv[src[t]] * ew[t] * dinv[dst[t]];
}

// ---------------------------------------------------------------------------
// Fused GCN-layer GEMM on V_WMMA_F32_16X16X4_F32.
//   in      : [N,128] previous aggregation (or x for layer 1)
//   If RELU : A-element = relu(in + bias)   (applies previous layer's bias+act)
//   C [N,128]   = A @ W                      (message features, read by scatter)
//   Agg[N,128]  = C * dinv(row)^2            (self-loop term, scatter target)
// Block = 256 threads (8 waves); each wave: one 16-row tile x eight 16-col
// tiles (64 acc VGPRs). W staged in 64KB LDS via async global->LDS copy.
// VGPR layouts per ISA 7.12.2 (32-bit):
//   A 16x4 : lanes 0-15 -> {K=k,k+1}, lanes 16-31 -> {K=k+2,k+3}, M = lane&15
//   B 4x16 : VGPR j holds row K = k + 2*(lane>>4) + j, N = lane&15
//   C/D    : VGPR v -> M = v + 8*(lane>>4), N = lane&15
template <bool RELU>
__global__ void gemm128_wmma(const float* __restrict__ A,
                             const float* __restrict__ W,
                             const float* __restrict__ bias,   // prev-layer bias
                             const float* __restrict__ dinv,
                             float* __restrict__ C,
                             float* __restrict__ Agg, int nRows) {
    __shared__ float Ws[NFEAT * NFEAT];        // 64 KB

    const float4* __restrict__ Wv = (const float4*)W;
    float4* Wsv = (float4*)Ws;
#ifdef ASYNC_LDS_COPY
    for (int i = threadIdx.x; i < NFEAT * NFEAT / 4; i += 256) {
        __builtin_amdgcn_global_load_async_to_lds_b128(
            (v4i*)(Wv + i), (v4i*)(Wsv + i), 0, 0);
    }
    __builtin_amdgcn_s_wait_asynccnt(0);
#else
    for (int i = threadIdx.x; i < NFEAT * NFEAT / 4; i += 256) Wsv[i] = Wv[i];
#endif
    __syncthreads();

    const int wave    = threadIdx.x >> 5;      // uniform per wave
    const int lane    = threadIdx.x & 31;
    const int rowTile = blockIdx.x * 8 + wave;
    if (rowTile >= (nRows >> 4)) return;       // wave-uniform exit: EXEC all-1s

    const int row   = rowTile << 4;
    const int mrow  = row + (lane & 15);
    const int khalf = lane >> 4;               // 0 or 1
    const int ncolL = lane & 15;

    const float* __restrict__ Arow = A + (size_t)mrow * NFEAT;
    __builtin_prefetch(Arow, 0, 0);            // global_prefetch_b8 (512B row)
    __builtin_prefetch(Arow + 64, 0, 0);

    v8f acc[8];
    #pragma unroll
    for (int t = 0; t < 8; ++t)
        #pragma unroll
        for (int j = 0; j < 8; ++j) acc[t][j] = 0.0f;

    for (int k = 0; k < NFEAT; k += 4) {
        const int kb = k + 2 * khalf;
        float2 av = *(const float2*)&Arow[kb];
        if (RELU) {
            const float2 bv = *(const float2*)&bias[kb];
            av.x = fmaxf(av.x + bv.x, 0.0f);
            av.y = fmaxf(av.y + bv.y, 0.0f);
        }
        v2f a; a[0] = av.x; a[1] = av.y;
        #pragma unroll
        for (int t = 0; t < 8; ++t) {
            const int ncol = (t << 4) + ncolL;
            v2f b;
            b[0] = Ws[kb * NFEAT + ncol];
            b[1] = Ws[(kb + 1) * NFEAT + ncol];
            acc[t] = __builtin_amdgcn_wmma_f32_16x16x4_f32(
                false, a, false, b, (short)0, acc[t], false, false);
        }
    }

    // Epilogue: C = hw ; Agg = hw * dinv^2 (self-loop contribution + zero-init)
    const int mbase = row + 8 * khalf;
    float dv2[8];
    #pragma unroll
    for (int v = 0; v < 8; ++v) {
        const float dv = dinv[mbase + v];
        dv2[v] = dv * dv;
    }
    #pragma unroll
    for (int t = 0; t < 8; ++t) {
        const int ncol = (t << 4) + ncolL;
        #pragma unroll
        for (int v = 0; v < 8; ++v) {
            const float val = acc[t][v];
            const size_t idx = (size_t)(mbase + v) * NFEAT + ncol;
            C[idx]   = val;
            Agg[idx] = val * dv2[v];
        }
    }
}

// ---------------------------------------------------------------------------
// One wave per edge: lane l handles features 4l..4l+3. Working set (~51MB)
// is L2-resident (192MB L2), so atomics stay on-chip.
__global__ void scatter_edges(const int* __restrict__ src, const int* __restrict__ dst,
                              const float* __restrict__ norm, const float* __restrict__ h,
                              float* __restrict__ agg, int nEdges) {
    const int t = blockIdx.x * blockDim.x + threadIdx.x;
    const int e = t >> 5;
    const int lane = t & 31;
    if (e >= nEdges) return;
    const int   s = src[e];
    const int   d = dst[e];
    const float w = norm[e];
    const float4 v = ((const float4*)&h[(size_t)s * NFEAT])[lane];
    float* ar = &agg[(size_t)d * NFEAT + 4 * lane];
    atomicAdd(ar + 0, v.x * w);
    atomicAdd(ar + 1, v.y * w);
    atomicAdd(ar + 2, v.z * w);
    atomicAdd(ar + 3, v.w * w);
}

// ---------------------------------------------------------------------------
// Sorted-batch mean-pool partials with fused layer-3 bias+relu.
// Block handles 128 consecutive nodes, thread = feature. Runs accumulate in
// registers; one atomic per graph run per thread.
__global__ void pool_kernel(const float* __restrict__ h, const float* __restrict__ bias,
                            const int* __restrict__ batch,
                            float* __restrict__ sums, float* __restrict__ cnts, int n) {
    const int f  = threadIdx.x;                // 0..127
    const int i0 = blockIdx.x * 128;
    const int i1 = min(i0 + 128, n);
    if (i0 >= n) return;
    const float bf = bias[f];
    int   cur = batch[i0];
    float acc = 0.0f;
    int   cl  = 0;
    for (int i = i0; i < i1; ++i) {
        const int g = batch[i];
        if (g != cur) {
            atomicAdd(&sums[(size_t)cur * NFEAT + f], acc);
            if (f == 0) atomicAdd(&cnts[cur], (float)cl);
            acc = 0.0f; cl = 0; cur = g;
        }
        acc += fmaxf(h[(size_t)i * NFEAT + f] + bf, 0.0f);
        ++cl;
    }
    atomicAdd(&sums[(size_t)cur * NFEAT + f], acc);
    if (f == 0) atomicAdd(&cnts[cur], (float)cl);
}

// One block per graph: g = sums/cnt; h1 = relu(g@Wl1+bl1); logits = h1@Wl2+bl2;
// out = log_softmax(logits)
__global__ void mlp_head(const float* __restrict__ sums, const float* __restrict__ cnts,
                         const float* __restrict__ Wl1, const float* __restrict__ bl1,
                         const float* __restrict__ Wl2, const float* __restrict__ bl2,
                         float* __restrict__ out) {
    __shared__ float gs[NFEAT];
    __shared__ float h1[NFEAT];
    __shared__ float lg[NCLS];
    const int r = blockIdx.x;
    const int f = threadIdx.x;

    const float denom = fmaxf(cnts[r], 1.0f);
    gs[f] = sums[(size_t)r * NFEAT + f] / denom;
    __syncthreads();

    float acc = bl1[f];
    for (int k = 0; k < NFEAT; ++k) acc += gs[k] * Wl1[k * NFEAT + f];
    h1[f] = fmaxf(acc, 0.0f);
    __syncthreads();

    if (f < NCLS) {
        float a = bl2[f];
        for (int k = 0; k < NFEAT; ++k) a += h1[k] * Wl2[k * NCLS + f];
        lg[f] = a;
    }
    __syncthreads();

    if (f == 0) {
        float m = lg[0];
        for (int c = 1; c < NCLS; ++c) m = fmaxf(m, lg[c]);
        float s = 0.0f;
        for (int c = 0; c < NCLS; ++c) s += expf(lg[c] - m);
        const float lse = m + logf(s);
        for (int c = 0; c < NCLS; ++c) out[(size_t)r * NCLS + c] = lg[c] - lse;
    }
}

// ---------------------------------------------------------------------------
extern "C" void kernel_launch(void* const* d_in, const int* in_sizes, int n_in,
                              void* d_out, int out_size, void* d_ws, size_t ws_size,
                              hipStream_t stream) {
    const float* x     = (const float*)d_in[0];
    const int*   ei    = (const int*)d_in[1];
    const float* ew    = (const float*)d_in[2];
    const int*   batch = (const int*)d_in[3];
    const float* W1 = (const float*)d_in[4];   const float* b1 = (const float*)d_in[5];
    const float* W2 = (const float*)d_in[6];   const float* b2 = (const float*)d_in[7];
    const float* W3 = (const float*)d_in[8];   const float* b3 = (const float*)d_in[9];
    const float* Wl1 = (const float*)d_in[10]; const float* bl1 = (const float*)d_in[11];
    const float* Wl2 = (const float*)d_in[12]; const float* bl2 = (const float*)d_in[13];
    float* out = (float*)d_out;

    const int N = in_sizes[0] / NFEAT;         // 100000
    const int E = in_sizes[2];                 // 1600000
    const int* src = ei;
    const int* dst = ei + E;

    // workspace layout (512B aligned slices)
    char* ws = (char*)d_ws;
    size_t off = 0;
    auto alloc = [&](size_t bytes) {
        void* p = ws + off;
        off = (off + bytes + 511) & ~(size_t)511;
        return p;
    };
    float* dinv = (float*)alloc((size_t)N * 4);
    float* norm = (float*)alloc((size_t)E * 4);
    float* bufA = (float*)alloc((size_t)N * NFEAT * 4);   // hw = A @ W
    float* bufB = (float*)alloc((size_t)N * NFEAT * 4);   // aggregation / next input
    float* sums = (float*)alloc((size_t)NGRAPH * NFEAT * 4);
    float* cnts = (float*)alloc((size_t)NGRAPH * 4);
    (void)ws_size; (void)n_in; (void)out_size;

    (void)hipMemsetAsync(dinv, 0, (size_t)N * 4, stream);
    (void)hipMemsetAsync(sums, 0, (size_t)NGRAPH * NFEAT * 4, stream);
    (void)hipMemsetAsync(cnts, 0, (size_t)NGRAPH * 4, stream);

    // GCN symmetric normalization
    deg_kernel  <<<(E + 255) / 256, 256, 0, stream>>>(dst, ew, dinv, E);
    rsqrt_kernel<<<(N + 255) / 256, 256, 0, stream>>>(dinv, N);
    norm_kernel <<<(E + 255) / 256, 256, 0, stream>>>(src, dst, ew, dinv, norm, E);

    const int gemmBlk = ((N >> 4) + 7) / 8;                    // 8 row-tiles/block
    const int scatBlk = (int)(((size_t)E * 32 + 255) / 256);   // 1 wave per edge
    const int poolBlk = (N + 127) / 128;

    // layer 1: A = x (no activation on load)
    gemm128_wmma<false><<<gemmBlk, 256, 0, stream>>>(x, W1, nullptr, dinv, bufA, bufB, N);
    scatter_edges<<<scatBlk, 256, 0, stream>>>(src, dst, norm, bufA, bufB, E);
    // layer 2: A = relu(agg1 + b1)
    gemm128_wmma<true><<<gemmBlk, 256, 0, stream>>>(bufB, W2, b1, dinv, bufA, bufB, N);
    scatter_edges<<<scatBlk, 256, 0, stream>>>(src, dst, norm, bufA, bufB, E);
    // layer 3: A = relu(agg2 + b2)
    gemm128_wmma<true><<<gemmBlk, 256, 0, stream>>>(bufB, W3, b2, dinv, bufA, bufB, N);
    scatter_edges<<<scatBlk, 256, 0, stream>>>(src, dst, norm, bufA, bufB, E);

    // pool applies relu(agg3 + b3), then mean; head + log_softmax
    pool_kernel<<<poolBlk, 128, 0, stream>>>(bufB, b3, batch, sums, cnts, N);
    mlp_head   <<<NGRAPH, 128, 0, stream>>>(sums, cnts, Wl1, bl1, Wl2, bl2, out);
}